// GCN_37512244363443
// MI455X (gfx1250) — compile-verified
//
#include <hip/hip_runtime.h>
#include <math.h>

typedef float v2f __attribute__((ext_vector_type(2)));
typedef float v8f __attribute__((ext_vector_type(8)));

#define IN_CH 128
#define HID_CH 128
#define OUT_CH 40

// ---------------------------------------------------------------------------
// GEMM1: H[n,128] = X[n,128] @ W[128,128]   (fp32 WMMA 16x16x4)
// Block = 256 threads = 8 waves; block covers 16 rows, wave w covers cols
// [16w,16w+16). K-loop: 32 x V_WMMA_F32_16X16X4_F32.
// ---------------------------------------------------------------------------
__global__ __launch_bounds__(256) void gemm1_kernel(const float* __restrict__ X,
                                                    const float* __restrict__ W,
                                                    float* __restrict__ H, int n) {
  const int wave = threadIdx.x >> 5;
  const int lane = threadIdx.x & 31;
  const int half = lane >> 4;      // 0: K pair {0,1}, 1: K pair {2,3}
  const int l    = lane & 15;
  const int row_base = blockIdx.x * 16;
  const int col = (wave << 4) + l;             // 0..127

  int arow_i = row_base + l;
  if (arow_i >= n) arow_i = n - 1;             // clamp (store is guarded)
  const float* __restrict__ arow = X + (size_t)arow_i * IN_CH;

  v8f acc = {};
  for (int k = 0; k < IN_CH; k += 4) {
    const int ka = k + (half << 1);
    v2f a, b;
    a.x = arow[ka];
    a.y = arow[ka + 1];
    b.x = W[(size_t)ka * HID_CH + col];
    b.y = W[(size_t)(ka + 1) * HID_CH + col];
    acc = __builtin_amdgcn_wmma_f32_16x16x4_f32(false, a, false, b,
                                                (short)0, acc, false, false);
  }

  float* __restrict__ crow = H + (size_t)(row_base + (half << 3)) * HID_CH + col;
  if (row_base + 16 <= n) {            // full tile: branch-free coalesced stores
#pragma unroll
    for (int v = 0; v < 8; ++v) crow[(size_t)v * HID_CH] = acc[v];
  } else {                             // ragged tail (not taken for n%16==0)
#pragma unroll
    for (int v = 0; v < 8; ++v) {
      if (row_base + (half << 3) + v < n) crow[(size_t)v * HID_CH] = acc[v];
    }
  }
}

// ---------------------------------------------------------------------------
// GEMM2: H[n,40] = A[n,128] @ W[128,40]   (fp32 WMMA, 3 col tiles, masked)
// Block = 96 threads = 3 waves covering cols 0..47 (cols >= 40 masked).
// ---------------------------------------------------------------------------
__global__ __launch_bounds__(96) void gemm2_kernel(const float* __restrict__ A,
                                                   const float* __restrict__ W,
                                                   float* __restrict__ H, int n) {
  const int wave = threadIdx.x >> 5;           // 0..2
  const int lane = threadIdx.x & 31;
  const int half = lane >> 4;
  const int l    = lane & 15;
  const int row_base = blockIdx.x * 16;
  const int col = (wave << 4) + l;             // 0..47
  const bool colok = (col < OUT_CH);

  int arow_i = row_base + l;
  if (arow_i >= n) arow_i = n - 1;
  const float* __restrict__ arow = A + (size_t)arow_i * HID_CH;

  v8f acc = {};
  for (int k = 0; k < HID_CH; k += 4) {
    const int ka = k + (half << 1);
    v2f a, b;
    a.x = arow[ka];
    a.y = arow[ka + 1];
    b.x = colok ? W[(size_t)ka * OUT_CH + col] : 0.0f;
    b.y = colok ? W[(size_t)(ka + 1) * OUT_CH + col] : 0.0f;
    acc = __builtin_amdgcn_wmma_f32_16x16x4_f32(false, a, false, b,
                                                (short)0, acc, false, false);
  }

  float* __restrict__ crow = H + (size_t)(row_base + (half << 3)) * OUT_CH + col;
  if (colok) {
    if (row_base + 16 <= n) {          // full tile fast path
#pragma unroll
      for (int v = 0; v < 8; ++v) crow[(size_t)v * OUT_CH] = acc[v];
    } else {
#pragma unroll
      for (int v = 0; v < 8; ++v) {
        if (row_base + (half << 3) + v < n) crow[(size_t)v * OUT_CH] = acc[v];
      }
    }
  }
}

// ---------------------------------------------------------------------------
// Edge aggregation, layer 1: AGG[dst] += H[src], 128 ch, one wave per edge.
// float4 gather (L2-resident) + 4 global_atomic_add_f32 per lane.
// ---------------------------------------------------------------------------
__global__ __launch_bounds__(256) void scatter1_kernel(const float* __restrict__ H,
                                                       const int* __restrict__ src,
                                                       const int* __restrict__ dst,
                                                       float* __restrict__ AGG,
                                                       int n_edges) {
  const int e = blockIdx.x * 8 + (threadIdx.x >> 5);
  if (e >= n_edges) return;
  const int lane = threadIdx.x & 31;
  const int s = src[e];
  const int d = dst[e];
  const float4 v = ((const float4*)(H + (size_t)s * HID_CH))[lane];
  float* out = AGG + (size_t)d * HID_CH + lane * 4;
  atomicAdd(out + 0, v.x);
  atomicAdd(out + 1, v.y);
  atomicAdd(out + 2, v.z);
  atomicAdd(out + 3, v.w);
}

// ---------------------------------------------------------------------------
// Edge aggregation, layer 2: OUT[dst] += H[src], 40 ch, one wave per edge.
// ---------------------------------------------------------------------------
__global__ __launch_bounds__(256) void scatter2_kernel(const float* __restrict__ H,
                                                       const int* __restrict__ src,
                                                       const int* __restrict__ dst,
                                                       float* __restrict__ OUT,
                                                       int n_edges) {
  const int e = blockIdx.x * 8 + (threadIdx.x >> 5);
  if (e >= n_edges) return;
  const int lane = threadIdx.x & 31;
  const int s = src[e];
  const int d = dst[e];
  if (lane < OUT_CH / 2) {
    const float2 v = ((const float2*)(H + (size_t)s * OUT_CH))[lane];
    float* out = OUT + (size_t)d * OUT_CH + lane * 2;
    atomicAdd(out + 0, v.x);
    atomicAdd(out + 1, v.y);
  }
}

// ---------------------------------------------------------------------------
// agg = relu(agg + b1)  (fused epilogue of layer 1)
// ---------------------------------------------------------------------------
__global__ __launch_bounds__(256) void bias_relu_kernel(float* __restrict__ A,
                                                        const float* __restrict__ b1,
                                                        long long total) {
  long long i = (long long)blockIdx.x * blockDim.x + threadIdx.x;
  const long long stride = (long long)gridDim.x * blockDim.x;
  for (; i < total; i += stride) {
    const float v = A[i] + b1[i & (HID_CH - 1)];
    A[i] = v > 0.0f ? v : 0.0f;
  }
}

// ---------------------------------------------------------------------------
// out = log_softmax(out + b2) per row of 40, one wave32 per row.
// lane covers ch {lane, lane+32 (lane<8)}; shuffle reductions for max / sum.
// ---------------------------------------------------------------------------
__global__ __launch_bounds__(256) void bias_logsoftmax_kernel(float* __restrict__ O,
                                                              const float* __restrict__ b2,
                                                              int n) {
  const int row = blockIdx.x * 8 + (threadIdx.x >> 5);
  if (row >= n) return;
  const int lane = threadIdx.x & 31;
  float* o = O + (size_t)row * OUT_CH;

  const float a = o[lane] + b2[lane];
  const bool hi = (lane < OUT_CH - 32);  // lane < 8
  const float b = hi ? (o[lane + 32] + b2[lane + 32]) : -INFINITY;

  float m = fmaxf(a, b);
  for (int off = 16; off > 0; off >>= 1) m = fmaxf(m, __shfl_xor(m, off, 32));
  float s = expf(a - m) + (hi ? expf(b - m) : 0.0f);
  for (int off = 16; off > 0; off >>= 1) s += __shfl_xor(s, off, 32);
  const float lse = m + logf(s);

  o[lane] = a - lse;
  if (hi) o[lane + 32] = b - lse;
}

// ---------------------------------------------------------------------------
__global__ __launch_bounds__(256) void zero_kernel(float4* __restrict__ p, long long n4) {
  long long i = (long long)blockIdx.x * blockDim.x + threadIdx.x;
  const long long stride = (long long)gridDim.x * blockDim.x;
  const float4 z = make_float4(0.0f, 0.0f, 0.0f, 0.0f);
  for (; i < n4; i += stride) p[i] = z;
}

// ---------------------------------------------------------------------------
extern "C" void kernel_launch(void* const* d_in, const int* in_sizes, int n_in,
                              void* d_out, int out_size, void* d_ws, size_t ws_size,
                              hipStream_t stream) {
  const float* x  = (const float*)d_in[0];
  const int*   ei = (const int*)d_in[1];
  const float* W1 = (const float*)d_in[2];
  const float* b1 = (const float*)d_in[3];
  const float* W2 = (const float*)d_in[4];
  const float* b2 = (const float*)d_in[5];

  const int n       = in_sizes[0] / IN_CH;
  const int n_edges = in_sizes[1] / 2;
  const int* src = ei;
  const int* dst = ei + n_edges;

  // Workspace: h1 [n*128] | agg1 [n*128]; h2 [n*40] reuses h1's slot.
  float* h1   = (float*)d_ws;
  float* agg1 = h1 + (size_t)n * HID_CH;
  float* h2   = h1;
  float* out  = (float*)d_out;

  zero_kernel<<<2048, 256, 0, stream>>>((float4*)agg1, (long long)n * HID_CH / 4);
  zero_kernel<<<2048, 256, 0, stream>>>((float4*)out,  (long long)n * OUT_CH / 4);

  // Layer 1: dense WMMA GEMM, then L2-resident gather + f32 atomic scatter
  gemm1_kernel<<<(n + 15) / 16, 256, 0, stream>>>(x, W1, h1, n);
  scatter1_kernel<<<(n_edges + 7) / 8, 256, 0, stream>>>(h1, src, dst, agg1, n_edges);
  bias_relu_kernel<<<2048, 256, 0, stream>>>(agg1, b1, (long long)n * HID_CH);

  // Layer 2
  gemm2_kernel<<<(n + 15) / 16, 96, 0, stream>>>(agg1, W2, h2, n);
  scatter2_kernel<<<(n_edges + 7) / 8, 256, 0, stream>>>(h2, src, dst, out, n_edges);
  bias_logsoftmax_kernel<<<(n + 7) / 8, 256, 0, stream>>>(out, b2, n);
}